// PredictYolov1_87325275062282
// MI455X (gfx1250) — compile-verified
//
#include <hip/hip_runtime.h>
#include <hip/hip_bf16.h>

// ---------------- problem constants ----------------
#define GS      7                       // grid size
#define NB      2                       // boxes per cell
#define NC      20                      // classes
#define NBATCH  128
#define NBOX    (NBATCH * GS * GS * NB) // 12544
#define NWORDS  (NBOX / 32)             // 392 32-col words per row
#define RTILES  (NBOX / 16)             // 784 16-row tiles
#define MSORT   16384                   // next pow2 >= NBOX
#define INVALID_GRP 0xFFFFFFFFu

typedef float v2f_t __attribute__((ext_vector_type(2)));
typedef float v8f_t __attribute__((ext_vector_type(8)));

#if __has_builtin(__builtin_amdgcn_wmma_f32_16x16x4_f32)
#define USE_WMMA_F32_K4 1
#else
#define USE_WMMA_F32_K4 0
#endif

__device__ __forceinline__ float sigmoidf_(float x) {
    return 1.0f / (1.0f + __expf(-x));
}

// ---- CDNA5 async global->LDS copy (ASYNCcnt) -------------------------------
// Inline asm: portable across ROCm-7.2 clang-22 and upstream clang-23.
// VDST = LDS byte address VGPR, VADDR = 64-bit global address (GV mode).
__device__ __forceinline__ void async_copy_b128(void* lds_dst, const void* gsrc) {
    unsigned lds_off = (unsigned)(size_t)lds_dst; // low 32 bits of flat LDS addr
    asm volatile("global_load_async_to_lds_b128 %0, %1, off"
                 :: "v"(lds_off), "v"(gsrc) : "memory");
}

__device__ __forceinline__ void wait_asynccnt0() {
#if __has_builtin(__builtin_amdgcn_s_wait_asynccnt)
    __builtin_amdgcn_s_wait_asynccnt(0);
#else
    asm volatile("s_wait_asynccnt 0" ::: "memory");
#endif
}

// ---------------------------------------------------------------------------
// Kernel 1: decode. One thread per (batch, cell, bbox).
// p layout [B, S, S, 30]: ch 0..7 = 2x(x,y,w,h), 8..9 = conf, 10..29 = classes
// ---------------------------------------------------------------------------
__global__ void yolo_decode_kernel(const float* __restrict__ p,
                                   float* __restrict__ scores,
                                   float* __restrict__ boxes,
                                   int*   __restrict__ labs) {
    int n = blockIdx.x * blockDim.x + threadIdx.x;
    if (n >= NBOX) return;
    int nb   = n & 1;
    int cell = n >> 1;
    int gx   = cell % GS;
    int gy   = (cell / GS) % GS;
    int b    = cell / (GS * GS);
    const float* q = p + ((size_t)((b * GS + gy) * GS + gx)) * 30;

    float tx = q[nb * 4 + 0];
    float ty = q[nb * 4 + 1];
    float w  = q[nb * 4 + 2];
    float h  = q[nb * 4 + 3];
    float conf = sigmoidf_(q[8 + nb]);

    float cx = (sigmoidf_(tx) + (float)gx) * (1.0f / GS);
    float cy = (sigmoidf_(ty) + (float)gy) * (1.0f / GS);

    int best = 0;
    float bv = q[10];
    #pragma unroll
    for (int c = 1; c < NC; ++c) {
        float v = q[10 + c];
        if (v > bv) { bv = v; best = c; }
    }

    scores[n] = conf;
    boxes[4 * n + 0] = cx - 0.5f * w;
    boxes[4 * n + 1] = cy - 0.5f * h;
    boxes[4 * n + 2] = cx + 0.5f * w;
    boxes[4 * n + 3] = cy + 0.5f * h;
    labs[n] = best + 1;
}

// ---------------------------------------------------------------------------
// Kernel 2: single-workgroup bitonic sort, 16384 u64 keys LDS-resident
// (128 KB -- needs CDNA5's 320 KB LDS/WGP). Descending by key:
//   key = ordered(score if valid else -inf) << 32 | (idx + 1), pad = 0.
// Tie at -inf -> descending original index, matching stable-argsort+reverse.
// ---------------------------------------------------------------------------
__global__ void __launch_bounds__(1024, 1)
yolo_sort_kernel(const float* __restrict__ scores, int* __restrict__ order) {
    __shared__ unsigned long long keys[MSORT];
    const int tid = threadIdx.x;

    for (int i = tid; i < MSORT; i += 1024) {
        unsigned long long key = 0ull;
        if (i < NBOX) {
            float s = scores[i];
            unsigned u = 0u; // invalid -> -inf bucket
            if (s > 0.5f) u = __float_as_uint(s) | 0x80000000u;
            key = ((unsigned long long)u << 32) | (unsigned)(i + 1);
        }
        keys[i] = key;
    }

    for (unsigned k = 2; k <= MSORT; k <<= 1) {
        for (unsigned j = k >> 1; j > 0; j >>= 1) {
            __syncthreads();
            for (unsigned i = tid; i < MSORT; i += 1024) {
                unsigned l = i ^ j;
                if (l > i) {
                    unsigned long long a = keys[i];
                    unsigned long long c = keys[l];
                    bool up = ((i & k) == 0);        // descending overall
                    if ((a < c) == up) { keys[i] = c; keys[l] = a; }
                }
            }
        }
    }
    __syncthreads();

    for (int i = tid; i < NBOX; i += 1024)
        order[i] = (int)(unsigned)(keys[i] & 0xFFFFFFFFull) - 1;
}

// ---------------------------------------------------------------------------
// Kernel 3: gather sorted outputs + build compact NMS inputs (boxes, areas,
// group codes). d_out (floats): [idsb N][boxes 4N][labs N][scores N][keep N]
// ---------------------------------------------------------------------------
__global__ void yolo_gather_kernel(const float* __restrict__ scores,
                                   const float* __restrict__ boxes,
                                   const int*   __restrict__ labs,
                                   const int*   __restrict__ order,
                                   float*       __restrict__ out,
                                   float*       __restrict__ sboxes,
                                   unsigned*    __restrict__ sgrp,
                                   float*       __restrict__ sarea) {
    int r = blockIdx.x * blockDim.x + threadIdx.x;
    if (r >= NBOX) return;
    int idx = order[r];
    float s = scores[idx];
    int b   = idx / (GS * GS * NB);
    int lab = labs[idx];

    float l  = boxes[4 * idx + 0];
    float t  = boxes[4 * idx + 1];
    float rr = boxes[4 * idx + 2];
    float bb = boxes[4 * idx + 3];

    out[r]                = (float)b;        // idsb_s
    out[NBOX + 4 * r + 0] = l;               // boxes_s
    out[NBOX + 4 * r + 1] = t;
    out[NBOX + 4 * r + 2] = rr;
    out[NBOX + 4 * r + 3] = bb;
    out[5 * NBOX + r]     = (float)lab;      // labs_s
    out[6 * NBOX + r]     = s;               // scores_s (unmasked)

    sboxes[4 * r + 0] = l;
    sboxes[4 * r + 1] = t;
    sboxes[4 * r + 2] = rr;
    sboxes[4 * r + 3] = bb;
    sarea[r] = (rr - l) * (bb - t);
    sgrp[r]  = (s > 0.5f) ? (((unsigned)b << 8) | (unsigned)lab) : INVALID_GRP;
}

// ---------------------------------------------------------------------------
// Kernel 4: fully parallel suppression-matrix build. One wave per 16x32
// region: two V_WMMA_F32_16X16X4_F32 ops compute the union term
// area_i + area_j for the two 16x16 tiles (A rows = [area_i,1,0,0],
// B cols = [1,area_j,0,0]); lanes evaluate intersection + threshold for
// their 8 C-fragment pairs; wave32 ballots pack verdicts into full 32-bit
// words of sup_mat (exclusively owned -> plain stores, no atomics).
// C layout: VGPR r <-> rows r (lanes 0-15) / r+8 (lanes 16-31), N = lane&15.
// ---------------------------------------------------------------------------
__global__ void __launch_bounds__(256, 2)
pairsup_kernel(const float* __restrict__ sboxes,
               const unsigned* __restrict__ sgrp,
               const float* __restrict__ sarea,
               unsigned* __restrict__ supmat) {
    const int gwave  = (int)((blockIdx.x * 256u + threadIdx.x) >> 5);
    const int lane   = (int)(threadIdx.x & 31u);
    const int rowTile = gwave / NWORDS;
    const int wcol    = gwave % NWORDS;
    const int i0 = rowTile * 16;
    const int j0 = wcol * 32;
    const int half = lane >> 4;      // 0: tile rows 0..7, 1: rows 8..15
    const int ln   = lane & 15;      // column within 16-wide tile

    const float4* bvec = (const float4*)sboxes;

    // column data for this lane (N = ln), both 16-wide column tiles
    float4  cb0 = bvec[j0 + ln];
    float4  cb1 = bvec[j0 + 16 + ln];
    unsigned gj0 = sgrp[j0 + ln];
    unsigned gj1 = sgrp[j0 + 16 + ln];
    float   ca0 = sarea[j0 + ln];
    float   ca1 = sarea[j0 + 16 + ln];

    // row data: this lane covers rows M = 8*half + r, r = 0..7
    float4   rb[8];
    unsigned gr[8];
    float    ra[8];
    #pragma unroll
    for (int r = 0; r < 8; ++r) {
        int row = i0 + 8 * half + r;
        rb[r] = bvec[row];
        gr[r] = sgrp[row];
        ra[r] = sarea[row];
    }

    float sums0[8], sums1[8];
#if USE_WMMA_F32_K4
    // A (16x4): VGPR0 = K0(lanes0-15)/K2(lanes16-31), VGPR1 = K1/K3
    // B (4x16): VGPR0 = K0 row / K2 row,              VGPR1 = K1 / K3
    float a_lo = sarea[i0 + ln]; // unconditional in-bounds load, select below
    v2f_t A, B0, B1;
    A.x  = half ? 0.0f : a_lo;   A.y  = half ? 0.0f : 1.0f;
    B0.x = half ? 0.0f : 1.0f;   B0.y = half ? 0.0f : ca0;
    B1.x = half ? 0.0f : 1.0f;   B1.y = half ? 0.0f : ca1;
    v8f_t C0 = {0.f,0.f,0.f,0.f,0.f,0.f,0.f,0.f};
    v8f_t C1 = {0.f,0.f,0.f,0.f,0.f,0.f,0.f,0.f};
    C0 = __builtin_amdgcn_wmma_f32_16x16x4_f32(false, A, false, B0,
                                               (short)0, C0, false, false);
    C1 = __builtin_amdgcn_wmma_f32_16x16x4_f32(false, A, false, B1,
                                               (short)0, C1, false, false);
    #pragma unroll
    for (int r = 0; r < 8; ++r) { sums0[r] = C0[r]; sums1[r] = C1[r]; }
#else
    #pragma unroll
    for (int r = 0; r < 8; ++r) { sums0[r] = ra[r] + ca0; sums1[r] = ra[r] + ca1; }
#endif

    const int jj0 = j0 + ln;
    const int jj1 = j0 + 16 + ln;
    #pragma unroll
    for (int r = 0; r < 8; ++r) {
        int irow = i0 + 8 * half + r;
        float iw0 = fmaxf(fminf(rb[r].z, cb0.z) - fmaxf(rb[r].x, cb0.x), 0.0f);
        float ih0 = fmaxf(fminf(rb[r].w, cb0.w) - fmaxf(rb[r].y, cb0.y), 0.0f);
        float in0 = iw0 * ih0;
        float un0 = fmaxf(sums0[r] - in0, 1e-9f);
        bool  s0  = (gj0 != INVALID_GRP) && (gr[r] == gj0) &&
                    (jj0 > irow) && (in0 > 0.3f * un0);

        float iw1 = fmaxf(fminf(rb[r].z, cb1.z) - fmaxf(rb[r].x, cb1.x), 0.0f);
        float ih1 = fmaxf(fminf(rb[r].w, cb1.w) - fmaxf(rb[r].y, cb1.y), 0.0f);
        float in1 = iw1 * ih1;
        float un1 = fmaxf(sums1[r] - in1, 1e-9f);
        bool  s1  = (gj1 != INVALID_GRP) && (gr[r] == gj1) &&
                    (jj1 > irow) && (in1 > 0.3f * un1);

        // b0 bits[15:0] = row i0+r cols j0..j0+15 ; bits[31:16] = row i0+r+8
        unsigned b0 = (unsigned)__ballot(s0);
        unsigned b1 = (unsigned)__ballot(s1);
        if (lane == 0) {
            unsigned wlo = (b0 & 0xFFFFu) | (b1 << 16);
            unsigned whi = (b0 >> 16)     | (b1 & 0xFFFF0000u);
            supmat[(size_t)(i0 + r) * NWORDS + wcol]     = wlo;
            supmat[(size_t)(i0 + r + 8) * NWORDS + wcol] = whi;
        }
    }
}

// ---------------------------------------------------------------------------
// Kernel 5: serial greedy scan. All FLOPs precomputed; each kept box just
// ORs its 392-word sup_mat row (L2-resident) into the LDS bitmask.
// Group codes staged via async global->LDS; next row prefetched each step.
// ---------------------------------------------------------------------------
__global__ void __launch_bounds__(1024, 1)
nms_scan_kernel(const unsigned* __restrict__ supmat,
                const unsigned* __restrict__ sgrp,
                float* __restrict__ out) {
    __shared__ __align__(16) unsigned sg[NBOX];     // 50176 B
    __shared__ unsigned validw[NWORDS];
    __shared__ unsigned supw[NWORDS];
    __shared__ unsigned keepw[NWORDS];

    const int tid  = threadIdx.x;
    const int lane = tid & 31;
    const int wv   = tid >> 5;

    // async-stage group codes (3136 x b128)
    for (int c = tid; c < NBOX / 4; c += 1024)
        async_copy_b128(&sg[4 * c], sgrp + 4 * c);
    for (int w = tid; w < NWORDS; w += 1024) { supw[w] = 0u; keepw[w] = 0u; }
    wait_asynccnt0();
    __syncthreads();

    // validity bitmask via wave32 ballots (each word exclusively owned)
    for (int k = 0; k < (NBOX + 1023) / 1024; ++k) {
        int r = k * 1024 + tid;
        bool v = (r < NBOX) && (sg[r] != INVALID_GRP);
        unsigned m = (unsigned)__ballot(v);
        int base = k * 1024 + wv * 32;
        if (lane == 0 && base < NBOX) validw[base >> 5] = m;
    }
    __syncthreads();

    for (int i = 0; i < NBOX; ++i) {
        // hide L2 latency of the next row while deciding this one
        if (tid < NWORDS && i + 1 < NBOX)
            __builtin_prefetch(&supmat[(size_t)(i + 1) * NWORDS + tid], 0, 0);

        bool keep = (((validw[i >> 5] >> (i & 31)) & 1u) != 0u) &&
                    (((supw[i >> 5]   >> (i & 31)) & 1u) == 0u);
        if (tid == 0 && keep) keepw[i >> 5] |= (1u << (i & 31));
        if (keep) { // uniform across the workgroup
            const unsigned* row = supmat + (size_t)i * NWORDS;
            if (tid < NWORDS) supw[tid] |= row[tid]; // row i never sets bit i (j>i)
        }
        __syncthreads();
    }

    for (int r = tid; r < NBOX; r += 1024)
        out[7 * NBOX + r] = (float)((keepw[r >> 5] >> (r & 31)) & 1u);
}

// ---------------------------------------------------------------------------
extern "C" void kernel_launch(void* const* d_in, const int* in_sizes, int n_in,
                              void* d_out, int out_size, void* d_ws, size_t ws_size,
                              hipStream_t stream) {
    const float* p = (const float*)d_in[0];
    float* out = (float*)d_out;

    // ws layout (float units): scores N | boxes 4N | labs N | order N |
    // sboxes 4N | sgrp N | sarea N | supmat (NBOX*NWORDS u32 ~ 19.7 MB)
    float*    ws     = (float*)d_ws;
    float*    scores = ws;
    float*    boxes  = ws + NBOX;
    int*      labs   = (int*)(ws + 5 * NBOX);
    int*      order  = (int*)(ws + 6 * NBOX);
    float*    sboxes = ws + 7 * NBOX;           // 16B aligned
    unsigned* sgrp   = (unsigned*)(ws + 11 * NBOX); // 16B aligned
    float*    sarea  = ws + 12 * NBOX;
    unsigned* supmat = (unsigned*)(ws + 13 * NBOX);

    dim3 blk(256);
    dim3 grd((NBOX + 255) / 256);

    yolo_decode_kernel<<<grd, blk, 0, stream>>>(p, scores, boxes, labs);
    yolo_sort_kernel<<<1, 1024, 0, stream>>>(scores, order);
    yolo_gather_kernel<<<grd, blk, 0, stream>>>(scores, boxes, labs, order,
                                                out, sboxes, sgrp, sarea);
    // one wave per 16x32 region: 784*392 regions, 8 waves per 256-thr block
    pairsup_kernel<<<(RTILES * NWORDS) / 8, 256, 0, stream>>>(sboxes, sgrp,
                                                              sarea, supmat);
    nms_scan_kernel<<<1, 1024, 0, stream>>>(supmat, sgrp, out);
}